// HankelCovLayer_23167053595379
// MI455X (gfx1250) — compile-verified
//
#include <hip/hip_runtime.h>

// CDNA5 / gfx1250, wave32. Batched 8x8 complex covariance of Hankel snapshots
// via V_WMMA_F32_16X16X4_F32 Gram accumulation: G = [Yr;Yi]*[Yr;Yi]^T (16x16),
//   Re(Ry) = G[0:8,0:8] + G[8:16,8:16]
//   Im(Ry) = G[8:16,0:8] - G[0:8,8:16]
// For the f32 16x16x4 WMMA the A (16x4) and B (4x16) VGPR layouts are register-
// identical transposes of each other, so the same v2f is fed as both A and B.

typedef __attribute__((ext_vector_type(2))) float v2f;
typedef __attribute__((ext_vector_type(8))) float v8f;

namespace {
constexpr int kS = 256;          // subcarriers per channel row
constexpr int kOutPerB = 128;    // 8*8*2 floats
constexpr int kLWin = 245;       // L = 252 - 8 + 1
constexpr int kFullChunks = 61;  // k0 = 0,4,...,240 : all lane k-offsets <= 243 valid
}

__global__ __launch_bounds__(256) void hankel_cov16_wmma(
    const float* __restrict__ in, float* __restrict__ out, int batch) {
  const int lane = threadIdx.x & 31;
  const int wave = threadIdx.x >> 5;
  const int b0 = (blockIdx.x * 8 + wave) * 2;  // two batch elements per wave
  if (b0 >= batch) return;                     // wave-uniform exit

  // Row m of the stacked 16 x 245 matrix [Yr; Yi]:
  //   m in 0..7  -> real channel, Hankel row m   -> in[b][0][2 + m + k]
  //   m in 8..15 -> imag channel, Hankel row m-8 -> in[b][1][2 + (m-8) + k]
  const int m = lane & 15;
  const int half = lane >> 4;  // A/B f32 K layout: lanes 0-15 hold k0+{0,1}, 16-31 hold k0+{2,3}
  const size_t rowoff = (size_t)(m >> 3) * kS + 2 + (m & 7);
  const float* src0 = in + (size_t)b0 * 2 * kS + rowoff;
  const float* src1 = src0 + 2 * kS;

  v8f acc0 = {};
  v8f acc1 = {};

  // Main accumulation: 61 full K=4 chunks. Same registers as A and B -> Gram.
  int kA = half * 2;
#pragma unroll 4
  for (int it = 0; it < kFullChunks; ++it, kA += 4) {
    v2f a0, a1;
    a0.x = src0[kA];
    a0.y = src0[kA + 1];
    a1.x = src1[kA];
    a1.y = src1[kA + 1];
    acc0 = __builtin_amdgcn_wmma_f32_16x16x4_f32(false, a0, false, a0,
                                                 (short)0, acc0, false, false);
    acc1 = __builtin_amdgcn_wmma_f32_16x16x4_f32(false, a1, false, a1,
                                                 (short)0, acc1, false, false);
  }

  // Tail chunk k0 = 244: only k = 244 (lane-half 0, element 0) is valid.
  // Branch-free (select after in-bounds load) so EXEC stays all-1s for WMMA.
  {
    const float t0 = src0[244];
    const float t1 = src1[244];
    v2f a0, a1;
    a0.x = half ? 0.0f : t0;
    a0.y = 0.0f;
    a1.x = half ? 0.0f : t1;
    a1.y = 0.0f;
    acc0 = __builtin_amdgcn_wmma_f32_16x16x4_f32(false, a0, false, a0,
                                                 (short)0, acc0, false, false);
    acc1 = __builtin_amdgcn_wmma_f32_16x16x4_f32(false, a1, false, a1,
                                                 (short)0, acc1, false, false);
  }

  // D layout (f32 16x16): VGPR r, lanes 0-15 -> G[r][lane]; lanes 16-31 -> G[8+r][lane-16].
  // Quadrants of 8 lanes for VGPR r=i:
  //   Q0 (0-7)  = G[i][j]        Q1 (8-15)  = G[i][8+j]
  //   Q2 (16-23)= G[8+i][j]      Q3 (24-31) = G[8+i][8+j]
  //   Re[i][j] = Q0 + Q3   (written by lanes 0-7,  comp 0)
  //   Im[i][j] = Q2 - Q1   (written by lanes 8-15, comp 1)
  const float scale = 1.0f / (float)kLWin;
  const int j = lane & 7;
  const int q = lane >> 3;                              // quadrant 0..3
  const int srcLane = (q == 0) ? (24 + j) : (16 + j);   // Q3 for re, Q2 for im
  float* o0 = out + (size_t)b0 * kOutPerB;
  float* o1 = o0 + kOutPerB;

#pragma unroll
  for (int i = 0; i < 8; ++i) {
    const float v0 = acc0[i];
    const float v1 = acc1[i];
    const float p0 = __shfl(v0, srcLane, 32);
    const float p1 = __shfl(v1, srcLane, 32);
    const float r0 = (q == 0) ? (v0 + p0) : (p0 - v0);
    const float r1 = (q == 0) ? (v1 + p1) : (p1 - v1);
    if (lane < 16) {
      const int idx = i * 16 + j * 2 + q;  // q in {0,1} here: re/im component
      o0[idx] = r0 * scale;
      o1[idx] = r1 * scale;
    }
  }
}

extern "C" void kernel_launch(void* const* d_in, const int* in_sizes, int n_in,
                              void* d_out, int out_size, void* d_ws, size_t ws_size,
                              hipStream_t stream) {
  const float* in = (const float*)d_in[0];
  float* out = (float*)d_out;
  const int batch = in_sizes[0] / (2 * kS);        // 32768 for the reference
  const int waves = (batch + 1) / 2;               // 2 batch elements per wave
  const int blocks = (waves + 7) / 8;              // 8 waves (256 threads) per block
  hankel_cov16_wmma<<<blocks, 256, 0, stream>>>(in, out, batch);
}